// DGODE_78039555768355
// MI455X (gfx1250) — compile-verified
//
#include <hip/hip_runtime.h>
#include <hip/hip_bf16.h>
#include <math.h>

// ---------------------------------------------------------------------------
// DGODE on MI455X (gfx1250): fp32 WMMA (V_WMMA_F32_16X16X4_F32) everywhere,
// adj materialized once in workspace (64MB, L2-resident), RK4 orchestrated as
// a kernel sequence on `stream`.
// ---------------------------------------------------------------------------

typedef __attribute__((ext_vector_type(2))) float v2f;
typedef __attribute__((ext_vector_type(8))) float v8f;

#define NB      4096
#define DIN     1856
#define HD      128
#define HN_ELEM (NB * HD)           // 524288 floats per state buffer

static __device__ __forceinline__ v8f wmma_f32(v2f a, v2f b, v8f c) {
    return __builtin_amdgcn_wmma_f32_16x16x4_f32(
        /*neg_a=*/false, a, /*neg_b=*/false, b,
        /*c_mod=*/(short)0, c, /*reuse_a=*/false, /*reuse_b=*/false);
}

// ---------------------------------------------------------------------------
// Kernel 1: build normalized adjacency.  One block per row i; 256 threads,
// each handling 16 columns.  Row sum via LDS tree reduction.
// ---------------------------------------------------------------------------
__global__ __launch_bounds__(256) void adj_kernel(const int* __restrict__ spk,
                                                  const float* __restrict__ mm,
                                                  float* __restrict__ adj) {
    const int i = blockIdx.x;
    const int t = threadIdx.x;
    __shared__ float red[256];
    __shared__ float mi[3];
    __shared__ int   si;
    if (t == 0) si = spk[i];
    if (t < 3)  mi[t] = mm[i * 3 + t];
    __syncthreads();
    const int   s_i = si;
    const float mi0 = mi[0], mi1 = mi[1], mi2 = mi[2];

    float vals[16];
    float sum = 0.0f;
    #pragma unroll
    for (int c = 0; c < 16; ++c) {
        const int j = t + 256 * c;
        float v;
        if (j == i) {
            v = 1.0f;
        } else {
            const float td       = fabsf((float)(i - j));
            const float temporal = __expf(-0.1f * td);
            if (spk[j] == s_i) {
                v = 0.8f * temporal;
            } else {
                const float md = fabsf(mi0 - mm[j * 3 + 0]) +
                                 fabsf(mi1 - mm[j * 3 + 1]) +
                                 fabsf(mi2 - mm[j * 3 + 2]);
                v = 0.5f * temporal * (1.0f - md * (1.0f / 3.0f));
            }
        }
        vals[c] = v;
        sum += v;
    }
    red[t] = sum;
    __syncthreads();
    for (int s = 128; s > 0; s >>= 1) {
        if (t < s) red[t] += red[t + s];
        __syncthreads();
    }
    const float inv = 1.0f / (red[0] + 1e-8f);
    #pragma unroll
    for (int c = 0; c < 16; ++c)
        adj[(size_t)i * NB + t + 256 * c] = vals[c] * inv;
}

// ---------------------------------------------------------------------------
// Kernel 2: input projection h0 = features @ Wp + bp  (4096x1856 @ 1856x128).
// One wave per 16x16 output tile; K marched in steps of 4 via f32 WMMA.
// ---------------------------------------------------------------------------
__global__ __launch_bounds__(32) void proj_kernel(const float* __restrict__ feat,
                                                  const float* __restrict__ Wp,
                                                  const float* __restrict__ bp,
                                                  float* __restrict__ h) {
    const int mt   = blockIdx.x * 16;
    const int nt   = blockIdx.y * 16;
    const int lane = threadIdx.x;
    const int ln   = lane & 15;        // M for A-frag, N for B/C/D frags
    const int hi   = lane >> 4;
    const int kb   = hi * 2;           // K sub-offset per ISA A/B layout

    v8f c = {};
    for (int k = 0; k < DIN; k += 4) {
        v2f a, b;
        a.x = feat[(size_t)(mt + ln) * DIN + k + kb];
        a.y = feat[(size_t)(mt + ln) * DIN + k + kb + 1];
        b.x = Wp[(size_t)(k + kb) * HD + nt + ln];
        b.y = Wp[(size_t)(k + kb + 1) * HD + nt + ln];
        c = wmma_f32(a, b, c);
    }
    const float bias = bp[nt + ln];
    #pragma unroll
    for (int r = 0; r < 8; ++r)
        h[(size_t)(mt + r + 8 * hi) * HD + nt + ln] = c[r] + bias;
}

// ---------------------------------------------------------------------------
// Kernel 3: neighbor aggregation hn = adj @ h  (4096x4096 @ 4096x128).
// Block = 8 waves = one 16-row x 128-col strip.  adj k-chunks (16x64) staged
// in LDS (padded stride 68) and shared across all 8 column-tile waves; h is
// L2-resident and loaded directly.
// ---------------------------------------------------------------------------
__global__ __launch_bounds__(256) void agg_kernel(const float* __restrict__ adj,
                                                  const float* __restrict__ h,
                                                  float* __restrict__ hn) {
    __shared__ float lA[16 * 68];
    const int mt   = blockIdx.x * 16;
    const int t    = threadIdx.x;
    const int wave = t >> 5;
    const int lane = t & 31;
    const int ln   = lane & 15;
    const int hi   = lane >> 4;
    const int kb   = hi * 2;
    const int nt   = wave * 16;

    v8f c = {};
    for (int k0 = 0; k0 < NB; k0 += 64) {
        // Stage 16x64 adj tile: 1024 floats by 256 threads (4 each)
        #pragma unroll
        for (int i = 0; i < 4; ++i) {
            const int idx = t + 256 * i;
            const int r   = idx >> 6;
            const int cc  = idx & 63;
            lA[r * 68 + cc] = adj[(size_t)(mt + r) * NB + k0 + cc];
        }
        __syncthreads();
        #pragma unroll
        for (int kk = 0; kk < 64; kk += 4) {
            v2f a, b;
            a.x = lA[ln * 68 + kk + kb];
            a.y = lA[ln * 68 + kk + kb + 1];
            b.x = h[(size_t)(k0 + kk + kb) * HD + nt + ln];
            b.y = h[(size_t)(k0 + kk + kb + 1) * HD + nt + ln];
            c = wmma_f32(a, b, c);
        }
        __syncthreads();
    }
    #pragma unroll
    for (int r = 0; r < 8; ++r)
        hn[(size_t)(mt + r + 8 * hi) * HD + nt + ln] = c[r];
}

// ---------------------------------------------------------------------------
// Kernel 4: fused MLP  out = tanh([h|hn] @ W1 + b1) @ W2 + b2.
// Block = 8 waves = one 16-row strip.  Phase-1 result (z, 16x128) goes
// through LDS to re-shape into A-fragments for phase 2.
// ---------------------------------------------------------------------------
__global__ __launch_bounds__(256) void mlp_kernel(const float* __restrict__ hsrc,
                                                  const float* __restrict__ hn,
                                                  const float* __restrict__ W1,
                                                  const float* __restrict__ b1,
                                                  const float* __restrict__ W2,
                                                  const float* __restrict__ b2,
                                                  float* __restrict__ kout) {
    __shared__ float z[16 * 132];
    const int mt   = blockIdx.x * 16;
    const int t    = threadIdx.x;
    const int wave = t >> 5;
    const int lane = t & 31;
    const int ln   = lane & 15;
    const int hi   = lane >> 4;
    const int kb   = hi * 2;
    const int nt   = wave * 16;

    // Phase 1: t1 = [h | hn] @ W1  (K = 256, split over the two halves)
    v8f c = {};
    #pragma unroll
    for (int k = 0; k < HD; k += 4) {
        v2f a, b;
        a.x = hsrc[(size_t)(mt + ln) * HD + k + kb];
        a.y = hsrc[(size_t)(mt + ln) * HD + k + kb + 1];
        b.x = W1[(size_t)(k + kb) * HD + nt + ln];
        b.y = W1[(size_t)(k + kb + 1) * HD + nt + ln];
        c = wmma_f32(a, b, c);
    }
    #pragma unroll
    for (int k = 0; k < HD; k += 4) {
        v2f a, b;
        a.x = hn[(size_t)(mt + ln) * HD + k + kb];
        a.y = hn[(size_t)(mt + ln) * HD + k + kb + 1];
        b.x = W1[(size_t)(HD + k + kb) * HD + nt + ln];
        b.y = W1[(size_t)(HD + k + kb + 1) * HD + nt + ln];
        c = wmma_f32(a, b, c);
    }
    const float bias1 = b1[nt + ln];
    #pragma unroll
    for (int r = 0; r < 8; ++r)
        z[(r + 8 * hi) * 132 + nt + ln] = tanhf(c[r] + bias1);
    __syncthreads();

    // Phase 2: out = z @ W2 + b2  (K = 128, A-frags from LDS)
    v8f c2 = {};
    #pragma unroll
    for (int k = 0; k < HD; k += 4) {
        v2f a, b;
        a.x = z[ln * 132 + k + kb];
        a.y = z[ln * 132 + k + kb + 1];
        b.x = W2[(size_t)(k + kb) * HD + nt + ln];
        b.y = W2[(size_t)(k + kb + 1) * HD + nt + ln];
        c2 = wmma_f32(a, b, c2);
    }
    const float bias2 = b2[nt + ln];
    #pragma unroll
    for (int r = 0; r < 8; ++r)
        kout[(size_t)(mt + r + 8 * hi) * HD + nt + ln] = c2[r] + bias2;
}

// ---------------------------------------------------------------------------
// Elementwise helpers
// ---------------------------------------------------------------------------
__global__ __launch_bounds__(256) void axpy_kernel(const float* __restrict__ h,
                                                   const float* __restrict__ k,
                                                   float s, float* __restrict__ out) {
    const int i = blockIdx.x * 256 + threadIdx.x;
    out[i] = h[i] + s * k[i];
}

__global__ __launch_bounds__(256) void rk4_combine_kernel(float* __restrict__ h,
                                                          const float* __restrict__ k1,
                                                          const float* __restrict__ k2,
                                                          const float* __restrict__ k3,
                                                          const float* __restrict__ k4,
                                                          float s) {
    const int i = blockIdx.x * 256 + threadIdx.x;
    h[i] = h[i] + s * (k1[i] + 2.0f * k2[i] + 2.0f * k3[i] + k4[i]);
}

__global__ __launch_bounds__(256) void copy_kernel(const float* __restrict__ src,
                                                   float* __restrict__ dst) {
    const int i = blockIdx.x * 256 + threadIdx.x;
    dst[i] = src[i];
}

// ---------------------------------------------------------------------------
// Host orchestration
// ---------------------------------------------------------------------------
extern "C" void kernel_launch(void* const* d_in, const int* in_sizes, int n_in,
                              void* d_out, int out_size, void* d_ws, size_t ws_size,
                              hipStream_t stream) {
    (void)in_sizes; (void)n_in; (void)out_size; (void)ws_size;
    const float* feat = (const float*)d_in[0];
    const int*   spk  = (const int*)  d_in[1];
    const float* mm   = (const float*)d_in[2];
    const float* Wp   = (const float*)d_in[3];
    const float* bp   = (const float*)d_in[4];
    const float* W1   = (const float*)d_in[5];
    const float* b1   = (const float*)d_in[6];
    const float* W2   = (const float*)d_in[7];
    const float* b2   = (const float*)d_in[8];
    float* out = (float*)d_out;

    float* ws  = (float*)d_ws;
    float* adj = ws;                           // 4096*4096 = 16777216 floats (64MB)
    float* h   = adj + (size_t)NB * NB;        // state buffers, 524288 floats each
    float* hn  = h  + HN_ELEM;
    float* ht  = hn + HN_ELEM;
    float* k1  = ht + HN_ELEM;
    float* k2  = k1 + HN_ELEM;
    float* k3  = k2 + HN_ELEM;
    float* k4  = k3 + HN_ELEM;

    const int EW_BLOCKS = HN_ELEM / 256;       // 2048
    const float dt = 1.0f / 4.0f;

    adj_kernel<<<NB, 256, 0, stream>>>(spk, mm, adj);
    proj_kernel<<<dim3(NB / 16, HD / 16), 32, 0, stream>>>(feat, Wp, bp, h);

    for (int step = 0; step < 4; ++step) {
        // k1 = f(h)
        agg_kernel<<<NB / 16, 256, 0, stream>>>(adj, h, hn);
        mlp_kernel<<<NB / 16, 256, 0, stream>>>(h, hn, W1, b1, W2, b2, k1);
        // k2 = f(h + 0.5*dt*k1)
        axpy_kernel<<<EW_BLOCKS, 256, 0, stream>>>(h, k1, 0.5f * dt, ht);
        agg_kernel<<<NB / 16, 256, 0, stream>>>(adj, ht, hn);
        mlp_kernel<<<NB / 16, 256, 0, stream>>>(ht, hn, W1, b1, W2, b2, k2);
        // k3 = f(h + 0.5*dt*k2)
        axpy_kernel<<<EW_BLOCKS, 256, 0, stream>>>(h, k2, 0.5f * dt, ht);
        agg_kernel<<<NB / 16, 256, 0, stream>>>(adj, ht, hn);
        mlp_kernel<<<NB / 16, 256, 0, stream>>>(ht, hn, W1, b1, W2, b2, k3);
        // k4 = f(h + dt*k3)
        axpy_kernel<<<EW_BLOCKS, 256, 0, stream>>>(h, k3, dt, ht);
        agg_kernel<<<NB / 16, 256, 0, stream>>>(adj, ht, hn);
        mlp_kernel<<<NB / 16, 256, 0, stream>>>(ht, hn, W1, b1, W2, b2, k4);
        // h += dt/6 * (k1 + 2k2 + 2k3 + k4)
        rk4_combine_kernel<<<EW_BLOCKS, 256, 0, stream>>>(h, k1, k2, k3, k4, dt / 6.0f);
    }

    copy_kernel<<<EW_BLOCKS, 256, 0, stream>>>(h, out);
}